// MyModel_61933428413646
// MI455X (gfx1250) — compile-verified
//
#include <hip/hip_runtime.h>
#include <hip/hip_bf16.h>

// fp32 WMMA fragment types (wave32): A 16x4 -> 2 VGPR, B 4x16 -> 2 VGPR, C/D 16x16 -> 8 VGPR
typedef __attribute__((ext_vector_type(2))) float v2f;
typedef __attribute__((ext_vector_type(8))) float v8f;

#define NB_   16          // batch
#define CIN   768         // K
#define HW    3136        // 56*56
#define COUT  384         // N
#define KC    32          // K chunk staged in LDS
#define BM    64          // pixels per block tile (2 waves x 32M)
#define BN    128         // couts per block tile  (4 waves x 32N)
// K-pair-interleaved LDS layout: element (k,col) at [(k>>1)*pitch + col*2 + (k&1)].
// Pitch % 64 == 32 dwords -> the two half-wave row-pairs occupy disjoint 32-bank windows.
#define P2X   160         // Xs row-pair pitch: 2*BM + 32
#define P2W   288         // Ws row-pair pitch: 2*BN + 32
#define NKCH  (CIN / KC)        // 24 chunks
#define MTPI  (HW / BM)         // 49 M-tiles per image
#define NTIL  (COUT / BN)       // 3 N-tiles

__global__ void zero_out_kernel(float* out) { out[0] = 0.0f; }

static __device__ __forceinline__ v8f wmma4(v2f a, v2f b, v8f c) {
    return __builtin_amdgcn_wmma_f32_16x16x4_f32(false, a, false, b, (short)0, c,
                                                 false, false);
}

__global__ __launch_bounds__(256)
void conv1x1_two_path_maxdiff(const float* __restrict__ x,
                              const float* __restrict__ W,
                              const float* __restrict__ bias,
                              float* __restrict__ out)
{
    __shared__ float Xs[(KC / 2) * P2X];   // 16 row-pairs x 160 = 10240 B
    __shared__ float Ws[(KC / 2) * P2W];   // 16 row-pairs x 288 = 18432 B
    __shared__ unsigned int smax;

    const int t    = threadIdx.x;
    const int lane = t & 31;
    const int wave = t >> 5;
    const int half = lane >> 4;   // 0: lanes 0-15 (K={0,1}), 1: lanes 16-31 (K={2,3})
    const int lm   = lane & 15;

    // block -> (image, pixel tile, cout tile)
    const int nb  = blockIdx.x % NTIL;
    const int mb  = blockIdx.x / NTIL;
    const int img = mb / MTPI;
    const int mt  = mb % MTPI;
    const int p0    = mt * BM;     // pixel base within image (contiguous, K-major layout)
    const int n0blk = nb * BN;     // cout base for block

    // 8 waves = 2 M-waves x 4 N-waves; each wave owns a 32M x 32N register tile
    const int m0 = (wave & 1) * 32;
    const int n0 = (wave >> 1) * 32;

    const float* xb = x + (size_t)img * CIN * HW + p0;   // + k*HW + pix

    const int aCol = m0 + lm;      // A column (pixel within block tile)
    const int bCol = n0 + lm;      // B column (cout within block tile)

    // X staging: each thread owns one K-row-pair x 4 pixels
    const int srp = t >> 4;              // 0..15 row-pair
    const int spx = (t & 15) * 4;        // 0..60 pixel base

    v8f c1[4] = {{}, {}, {}, {}};
    v8f c2[4] = {{}, {}, {}, {}};

    for (int pass = 0; pass < 2; ++pass) {
        v8f acc[4] = {{}, {}, {}, {}};
        for (int kk = 0; kk < NKCH; ++kk) {
            const int kc = (pass == 0 ? kk : (NKCH - 1 - kk)) * KC;

            // ---- stage X chunk (KC x BM), K-pair interleaved: b128 loads, b64 LDS stores ----
            {
                const float* g0 = xb + (size_t)(kc + 2 * srp) * HW + spx;
                const float4 v = *reinterpret_cast<const float4*>(g0);       // row 2*srp
                const float4 u = *reinterpret_cast<const float4*>(g0 + HW);  // row 2*srp+1
                float* base = &Xs[srp * P2X + spx * 2];
                v2f p;
                p.x = v.x; p.y = u.x; *reinterpret_cast<v2f*>(base + 0) = p;
                p.x = v.y; p.y = u.y; *reinterpret_cast<v2f*>(base + 2) = p;
                p.x = v.z; p.y = u.z; *reinterpret_cast<v2f*>(base + 4) = p;
                p.x = v.w; p.y = u.w; *reinterpret_cast<v2f*>(base + 6) = p;
            }

            // ---- stage W chunk (BN x KC) transposed + interleaved: b128 loads, b64 stores ----
            #pragma unroll
            for (int i = 0; i < 4; ++i) {
                const int fi = t * 4 + i * 1024;        // 4096 floats total
                const int o  = fi >> 5;                 // 0..127 couts
                const int k  = fi & (KC - 1);           // 0,4,...,28
                const float4 v = *reinterpret_cast<const float4*>(
                    W + (size_t)(n0blk + o) * CIN + kc + k);
                const int kp = k >> 1;                  // even row-pair index
                v2f p;
                p.x = v.x; p.y = v.y;
                *reinterpret_cast<v2f*>(&Ws[(kp + 0) * P2W + o * 2]) = p;
                p.x = v.z; p.y = v.w;
                *reinterpret_cast<v2f*>(&Ws[(kp + 1) * P2W + o * 2]) = p;
            }

            // prefetch the next K-slab of x into cache (global_prefetch_b8)
            if (kk + 1 < NKCH) {
                const int kn = (pass == 0 ? (kk + 1) : (NKCH - 2 - kk)) * KC;
                __builtin_prefetch(xb + (size_t)kn * HW + (t & 31) * 64, 0, 1);
            }

            __syncthreads();

            // ---- 8 K-steps x 4 WMMAs (2x2 register blocking); fragments = single b64 loads ----
            #pragma unroll
            for (int k4 = 0; k4 < KC; k4 += 4) {
                const int ka = (k4 >> 1) + half;        // row-pair holding K={k4+rA, k4+rA+1}
                const float* ax = &Xs[ka * P2X + aCol * 2];
                const float* bx = &Ws[ka * P2W + bCol * 2];
                const v2f a0 = *reinterpret_cast<const v2f*>(ax);
                const v2f a1 = *reinterpret_cast<const v2f*>(ax + 32);   // col + 16
                const v2f b0 = *reinterpret_cast<const v2f*>(bx);
                const v2f b1 = *reinterpret_cast<const v2f*>(bx + 32);   // col + 16
                acc[0] = wmma4(a0, b0, acc[0]);
                acc[1] = wmma4(a0, b1, acc[1]);
                acc[2] = wmma4(a1, b0, acc[2]);
                acc[3] = wmma4(a1, b1, acc[3]);
            }

            __syncthreads();
        }
        #pragma unroll
        for (int i = 0; i < 4; ++i) {
            if (pass == 0) c1[i] = acc[i]; else c2[i] = acc[i];
        }
    }

    // Both paths add the same bias (as in the reference) before differencing.
    // C/D layout: the lane's N column is fixed across all 8 VGPRs of a tile.
    const float bb0 = bias[n0blk + bCol];        // N sub-tile j=0
    const float bb1 = bias[n0blk + bCol + 16];   // N sub-tile j=1
    float m = 0.0f;
    #pragma unroll
    for (int i = 0; i < 4; ++i) {
        const float bb = (i & 1) ? bb1 : bb0;    // acc index: [mi*2 + nj]
        #pragma unroll
        for (int v = 0; v < 8; ++v) {
            const float d = __builtin_fabsf((c1[i][v] + bb) - (c2[i][v] + bb));
            m = __builtin_fmaxf(m, d);
        }
    }

    // block max (diffs are >= 0, so uint compare == float compare)
    if (t == 0) smax = 0u;
    __syncthreads();
    atomicMax(&smax, __float_as_uint(m));
    __syncthreads();
    if (t == 0) atomicMax(reinterpret_cast<unsigned int*>(out), smax);
}

extern "C" void kernel_launch(void* const* d_in, const int* in_sizes, int n_in,
                              void* d_out, int out_size, void* d_ws, size_t ws_size,
                              hipStream_t stream) {
    const float* x  = (const float*)d_in[0];   // [16,768,56,56]
    const float* W  = (const float*)d_in[1];   // [384,768]
    const float* b  = (const float*)d_in[2];   // [384]
    float* out = (float*)d_out;                // scalar

    zero_out_kernel<<<1, 1, 0, stream>>>(out);

    const int grid = (NB_ * HW / BM) * NTIL;   // 784 * 3 = 2352 blocks
    conv1x1_two_path_maxdiff<<<grid, 256, 0, stream>>>(x, W, b, out);
}